// magnetics_82695300317659
// MI455X (gfx1250) — compile-verified
//
#include <hip/hip_runtime.h>

// ---------------------------------------------------------------------------
// Magnetics forward model (512x512x128) for MI455X / gfx1250.
//
// B = sum_z Re( ifft2( fft2(fftshift(M_z)) * fft2(fftshift(roll(P_z))) ) ) * zeta*dV
// (output-side fftshifts cancel pairwise; the z-sum is moved inside one ifft2).
//
// Every 512-point FFT runs on one wave32 via the four-step factorization
// 512 = 16*32, both small-DFT stages expressed as f32 GEMMs on
// V_WMMA_F32_16X16X4_F32.  Row passes use a real-input specialization
// (imaginary part identically zero -> stage 1 needs half the WMMAs).
// FFT staging is SoA so WMMA fragments load as ds_load_2addr_b32 pairs
// straight into the even-aligned VGPR pairs the matrix op consumes.
// DFT/twiddle tables are built once in global memory and pulled into LDS
// per block via GLOBAL_LOAD_ASYNC_TO_LDS_B128 (ASYNCcnt-tracked).
//
// HBM traffic ~= read M (128 MB) + write out (1 MB); every intermediate
// (row spectra, accumulator, tables) is L2-resident (192 MB L2).
// ---------------------------------------------------------------------------

#define NXD 512
#define NYD 512
#define NZD 128

constexpr int   WPB    = 4;          // waves (=FFTs) per block
constexpr int   TPB    = WPB * 32;   // 128 threads
constexpr int   BZ     = 8;          // z-slices per launch round
constexpr int   NTAB   = 3584;       // table floats: w32(2048) + w16(512) + tw(1024)
constexpr float TWO_PI = 6.283185307179586f;

typedef float v2f __attribute__((ext_vector_type(2)));
typedef float v8f __attribute__((ext_vector_type(8)));
typedef int   b128v __attribute__((vector_size(16)));  // async-LDS payload type

#if defined(__has_builtin)
#if __has_builtin(__builtin_amdgcn_global_load_async_to_lds_b128) && \
    __has_builtin(__builtin_amdgcn_s_wait_asynccnt)
#define HAS_ASYNC_LDS 1
#endif
#endif
#ifndef HAS_ASYNC_LDS
#define HAS_ASYNC_LDS 0
#endif

__device__ inline v8f z8() {
  v8f r;
#pragma unroll
  for (int i = 0; i < 8; ++i) r[i] = 0.0f;
  return r;
}

// Output position of result element (r, t) held by `lane` after stage 2:
// k1 = r + 8*hi, k2 = lo + 16*t, X-index = 32*k1 + k2.
__device__ inline int xidx(int r, int t, int lane) {
  return 32 * ((lane >> 4) * 8 + r) + (lane & 15) + 16 * t;
}

// --- One-time table build into workspace (SoA, forward sign: -2*pi*nk/N).
// Layout: [0,1024) W32r | [1024,2048) W32i | [2048,2304) W16r |
//         [2304,2560) W16i | [2560,3072) TWr | [3072,3584) TWi
__global__ void k_build_tables(float* __restrict__ g) {
  for (int i = threadIdx.x; i < 32 * 32; i += TPB) {
    int n = i >> 5, k = i & 31;
    float s, c;
    sincosf(-TWO_PI * (float)((n * k) & 31) * (1.0f / 32.0f), &s, &c);
    g[i] = c; g[1024 + i] = s;
  }
  for (int i = threadIdx.x; i < 16 * 16; i += TPB) {
    int n = i >> 4, k = i & 15;
    float s, c;
    sincosf(-TWO_PI * (float)((n * k) & 15) * (1.0f / 16.0f), &s, &c);
    g[2048 + i] = c; g[2304 + i] = s;
  }
  for (int i = threadIdx.x; i < 16 * 32; i += TPB) {
    int n1 = i >> 5, k2 = i & 31;
    float s, c;
    sincosf(-TWO_PI * (float)(n1 * k2) * (1.0f / 512.0f), &s, &c);
    g[2560 + i] = c; g[3072 + i] = s;
  }
}

// --- Pull tables global -> LDS.  Async-LDS path bypasses VGPRs and is
// tracked by ASYNCcnt; each wave waits on its own issued asyncs, the
// following __syncthreads() in the caller covers cross-wave visibility.
__device__ inline void load_tables(const float* __restrict__ g, float* sTab) {
#if HAS_ASYNC_LDS
  for (int i = threadIdx.x * 4; i < NTAB; i += TPB * 4) {
    __builtin_amdgcn_global_load_async_to_lds_b128(
        (__attribute__((address_space(1))) b128v*)(g + i),
        (__attribute__((address_space(3))) b128v*)(sTab + i), 0, 0);
  }
  __builtin_amdgcn_s_wait_asynccnt(0);
#else
  for (int i = threadIdx.x; i < NTAB; i += TPB) sTab[i] = g[i];
#endif
}

// ---------------------------------------------------------------------------
// One 512-point complex FFT on one wave32.
//   Stage 1: B[n1][k2] = sum_{n2<32} x[n1+16*n2] * W32^{n2*k2}   (16x32 GEMM, K=32)
//   Twiddle: B[n1][k2] *= W512^{n1*k2}
//   Stage 2: X[k2+32*k1] = sum_{n1<16} W16^{n1*k1} * B[n1][k2]   (16x32 GEMM, K=16)
// Complex GEMM = two accumulator chains (re, im), 4 WMMAs per K-chunk
// (2 when REAL_IN: the Ai terms vanish for purely real input).
// A fragment: lane L holds row M=(L&15), K pair {2*(L>>4), 2*(L>>4)+1}.
// B fragment: VGPR r holds row K=r+2*(L>>4), column N=(L&15)+16*tile.
// ---------------------------------------------------------------------------
template <bool INV, bool REAL_IN>
__device__ void fft512(const float* w32r, const float* w32i,
                       const float* w16r, const float* w16i,
                       const float* twr,  const float* twi,
                       const float* inRe, const float* inIm,
                       float* midRe, float* midIm,
                       float2* res, int lane) {
  const int lo = lane & 15, hi = lane >> 4;

  v8f cre0 = z8(), cim0 = z8(), cre1 = z8(), cim1 = z8();
#pragma unroll
  for (int kk = 0; kk < 8; ++kk) {  // K = 32 over n2, chunks of 4
    const int kb = 4 * kk + 2 * hi;
    const int ia0 = lo + 16 * kb, ia1 = lo + 16 * (kb + 1);
    float ar0 = inRe[ia0], ar1 = inRe[ia1];
    v2f Ar = {ar0, ar1};
    v2f Ai, nAi;
    if (!REAL_IN) {
      float ai0 = inIm[ia0], ai1 = inIm[ia1];
      Ai  = v2f{ ai0,  ai1};
      nAi = v2f{-ai0, -ai1};
    }
#pragma unroll
    for (int t = 0; t < 2; ++t) {
      const int col = lo + 16 * t;
      const int ib0 = kb * 32 + col, ib1 = (kb + 1) * 32 + col;
      float br0 = w32r[ib0], br1 = w32r[ib1];
      float bi0 = w32i[ib0], bi1 = w32i[ib1];
      if (INV) { bi0 = -bi0; bi1 = -bi1; }   // conjugate table for inverse
      v2f Br = {br0, br1};
      v2f Bi = {bi0, bi1};
      v8f& cre = t ? cre1 : cre0;
      v8f& cim = t ? cim1 : cim0;
      // Cre += Ar*Br - Ai*Bi ; Cim += Ar*Bi + Ai*Br
      cre = __builtin_amdgcn_wmma_f32_16x16x4_f32(false, Ar, false, Br, (short)0, cre, false, false);
      cim = __builtin_amdgcn_wmma_f32_16x16x4_f32(false, Ar, false, Bi, (short)0, cim, false, false);
      if (!REAL_IN) {
        cre = __builtin_amdgcn_wmma_f32_16x16x4_f32(false, nAi, false, Bi, (short)0, cre, false, false);
        cim = __builtin_amdgcn_wmma_f32_16x16x4_f32(false, Ai,  false, Br, (short)0, cim, false, false);
      }
    }
  }

  // Twiddle, spill stage-1 result to LDS as mid[n1*32 + k2] (SoA).
  // D layout: VGPR r -> row M = r + 8*hi, col N = lo + 16*t.
#pragma unroll
  for (int t = 0; t < 2; ++t) {
    v8f cre = t ? cre1 : cre0;
    v8f cim = t ? cim1 : cim0;
    const int col = lo + 16 * t;
#pragma unroll
    for (int r = 0; r < 8; ++r) {
      const int n1 = r + 8 * hi;
      const int iw = n1 * 32 + col;
      float wr = twr[iw];
      float wi = INV ? -twi[iw] : twi[iw];
      float vr = cre[r], vi = cim[r];
      midRe[iw] = vr * wr - vi * wi;
      midIm[iw] = vr * wi + vi * wr;
    }
  }
  // Same-wave LDS ops are in order on CDNA5; no cross-wave sharing here.

  cre0 = z8(); cim0 = z8(); cre1 = z8(); cim1 = z8();
#pragma unroll
  for (int kk = 0; kk < 4; ++kk) {  // K = 16 over n1
    const int kb = 4 * kk + 2 * hi;
    const int ia0 = kb * 16 + lo, ia1 = (kb + 1) * 16 + lo;
    float ar0 = w16r[ia0], ar1 = w16r[ia1];
    float ai0 = w16i[ia0], ai1 = w16i[ia1];
    if (INV) { ai0 = -ai0; ai1 = -ai1; }     // conjugate table for inverse
    v2f Ar = {ar0, ar1};
    v2f Ai = {ai0, ai1};
#pragma unroll
    for (int t = 0; t < 2; ++t) {
      const int col = lo + 16 * t;
      const int ib0 = kb * 32 + col, ib1 = (kb + 1) * 32 + col;
      float br0 = midRe[ib0], br1 = midRe[ib1];
      float bi0 = midIm[ib0], bi1 = midIm[ib1];
      v2f Br  = { br0,  br1};
      v2f Bi  = { bi0,  bi1};
      v2f nBi = {-bi0, -bi1};
      v8f& cre = t ? cre1 : cre0;
      v8f& cim = t ? cim1 : cim0;
      cre = __builtin_amdgcn_wmma_f32_16x16x4_f32(false, Ar, false, Br,  (short)0, cre, false, false);
      cre = __builtin_amdgcn_wmma_f32_16x16x4_f32(false, Ai, false, nBi, (short)0, cre, false, false);
      cim = __builtin_amdgcn_wmma_f32_16x16x4_f32(false, Ar, false, Bi,  (short)0, cim, false, false);
      cim = __builtin_amdgcn_wmma_f32_16x16x4_f32(false, Ai, false, Br,  (short)0, cim, false, false);
    }
  }
#pragma unroll
  for (int t = 0; t < 2; ++t) {
    v8f cre = t ? cre1 : cre0;
    v8f cim = t ? cim1 : cim0;
#pragma unroll
    for (int r = 0; r < 8; ++r) res[t * 8 + r] = make_float2(cre[r], cim[r]);
  }
}

#define FFT_SHARED                          \
  __shared__ alignas(16) float sTab[NTAB];  \
  __shared__ float sInRe[WPB][512];         \
  __shared__ float sInIm[WPB][512];         \
  __shared__ float sMidRe[WPB][512];        \
  __shared__ float sMidIm[WPB][512];        \
  const float* sW32r = sTab;                \
  const float* sW32i = sTab + 1024;         \
  const float* sW16r = sTab + 2048;         \
  const float* sW16i = sTab + 2304;         \
  const float* sTWr  = sTab + 2560;         \
  const float* sTWi  = sTab + 3072;

#define FFT_ARGS sW32r, sW32i, sW16r, sW16i, sTWr, sTWi

// --- Row FFT of fftshift(M_z): output row x is FFT_y of M[(x^256)][(y^256)][z].
// Real input: no imaginary plane is written or read.
__global__ __launch_bounds__(TPB) void k_fftrow_M(const float* __restrict__ M,
                                                  const float* __restrict__ tabs,
                                                  float2* __restrict__ rowM, int zbase) {
  FFT_SHARED
  load_tables(tabs, sTab);
  __syncthreads();
  const int lane = threadIdx.x & 31, wv = threadIdx.x >> 5;
  const int x = blockIdx.x * WPB + wv, j = blockIdx.y, z = zbase + j;
  const int xs = x ^ 256;
#pragma unroll
  for (int i = 0; i < 16; ++i) {
    int y = lane + 32 * i;
    sInRe[wv][y] = M[((size_t)xs * NYD + (y ^ 256)) * NZD + z];
  }
  float2 res[16];
  fft512<false, true>(FFT_ARGS, sInRe[wv], nullptr, sMidRe[wv], sMidIm[wv], res, lane);
  float2* dst = rowM + ((size_t)(j * NXD + x)) * NYD;
#pragma unroll
  for (int t = 0; t < 2; ++t)
#pragma unroll
    for (int r = 0; r < 8; ++r) dst[xidx(r, t, lane)] = res[t * 8 + r];
}

// --- Row FFT of fftshift(roll(P_z)): roll(center)+fftshift compose to index-1.
// P is analytic (and real): generated in registers, never read from memory.
__global__ __launch_bounds__(TPB) void k_fftrow_P(const float* __restrict__ h,
                                                  const float* __restrict__ dirs,
                                                  const float* __restrict__ tabs,
                                                  float2* __restrict__ rowP, int zbase) {
  FFT_SHARED
  load_tables(tabs, sTab);
  __syncthreads();
  const int lane = threadIdx.x & 31, wv = threadIdx.x >> 5;
  const int x = blockIdx.x * WPB + wv, j = blockIdx.y, z = zbase + j;
  const float Dx = h[0], Dy = h[1], Dz = h[2];
  float sI, cI, sA, cA, sI0, cI0, sA0, cA0;
  sincosf(dirs[0], &sI, &cI);
  sincosf(dirs[1], &sA, &cA);
  sincosf(dirs[2], &sI0, &cI0);
  sincosf(dirs[3], &sA0, &cA0);
  const float Z = Dz * (0.5f + (float)z);
  const float X = Dx * (float)(((x - 1) & 511) - 255);
  const float w1 = cI * sA, w2 = cI * cA;
#pragma unroll
  for (int i = 0; i < 16; ++i) {
    int y = lane + 32 * i;
    float Y = Dy * (float)(((y - 1) & 511) - 255);
    float r2 = X * X + Y * Y + Z * Z;
    float inv = 1.0f / (r2 * r2 * sqrtf(r2));  // 1 / r^5
    float px = ((2.f * X * X - Y * Y - Z * Z) * w1 + 3.f * X * Y * w2 + 3.f * X * Z * sI) * inv;
    float py = (3.f * X * Y * w1 + (2.f * Y * Y - X * X - Z * Z) * w2 + 3.f * Y * Z * sI) * inv;
    float pz = (3.f * X * Z * w1 + 3.f * Z * Y * w2 + (2.f * Z * Z - X * X - Y * Y) * sI) * inv;
    sInRe[wv][y] = px * cI0 * cA0 + py * cI0 * sA0 + pz * sI0;
  }
  float2 res[16];
  fft512<false, true>(FFT_ARGS, sInRe[wv], nullptr, sMidRe[wv], sMidIm[wv], res, lane);
  float2* dst = rowP + ((size_t)(j * NXD + x)) * NYD;
#pragma unroll
  for (int t = 0; t < 2; ++t)
#pragma unroll
    for (int r = 0; r < 8; ++r) dst[xidx(r, t, lane)] = res[t * 8 + r];
}

// --- Column FFT of both spectra + frequency-domain MAC into acc[j][ky][kx].
// Distinct (j, col) per block -> no races, deterministic accumulation.
__global__ __launch_bounds__(TPB) void k_fftcol_mac(const float2* __restrict__ rowM,
                                                    const float2* __restrict__ rowP,
                                                    const float* __restrict__ tabs,
                                                    float2* __restrict__ acc) {
  FFT_SHARED
  load_tables(tabs, sTab);
  __syncthreads();
  const int lane = threadIdx.x & 31, wv = threadIdx.x >> 5;
  const int col = blockIdx.x * WPB + wv, j = blockIdx.y;

  const float2* srcM = rowM + (size_t)j * NXD * NYD;
#pragma unroll
  for (int i = 0; i < 16; ++i) {
    int xx = lane + 32 * i;
    float2 v = srcM[(size_t)xx * NYD + col];
    sInRe[wv][xx] = v.x;
    sInIm[wv][xx] = v.y;
  }
  float2 resM[16];
  fft512<false, false>(FFT_ARGS, sInRe[wv], sInIm[wv], sMidRe[wv], sMidIm[wv], resM, lane);

  const float2* srcP = rowP + (size_t)j * NXD * NYD;
#pragma unroll
  for (int i = 0; i < 16; ++i) {
    int xx = lane + 32 * i;
    float2 v = srcP[(size_t)xx * NYD + col];
    sInRe[wv][xx] = v.x;
    sInIm[wv][xx] = v.y;
  }
  float2 resP[16];
  fft512<false, false>(FFT_ARGS, sInRe[wv], sInIm[wv], sMidRe[wv], sMidIm[wv], resP, lane);

  float2* a = acc + ((size_t)(j * NYD + col)) * NXD;
#pragma unroll
  for (int t = 0; t < 2; ++t)
#pragma unroll
    for (int r = 0; r < 8; ++r) {
      int kx = xidx(r, t, lane);
      float2 m = resM[t * 8 + r], p = resP[t * 8 + r];
      float2 o = a[kx];
      a[kx] = make_float2(o.x + m.x * p.x - m.y * p.y,
                          o.y + m.x * p.y + m.y * p.x);
    }
}

// --- Inverse FFT over kx (reducing the BZ accumulator slots on load).
__global__ __launch_bounds__(TPB) void k_ifft_rows(const float2* __restrict__ acc,
                                                   const float* __restrict__ tabs,
                                                   float2* __restrict__ tmpb) {
  FFT_SHARED
  load_tables(tabs, sTab);
  __syncthreads();
  const int lane = threadIdx.x & 31, wv = threadIdx.x >> 5;
  const int c = blockIdx.x * WPB + wv;  // ky
#pragma unroll
  for (int i = 0; i < 16; ++i) {
    int kx = lane + 32 * i;
    float sr = 0.0f, si = 0.0f;
#pragma unroll
    for (int j = 0; j < BZ; ++j) {
      float2 v = acc[((size_t)(j * NYD + c)) * NXD + kx];
      sr += v.x;
      si += v.y;
    }
    sInRe[wv][kx] = sr;
    sInIm[wv][kx] = si;
  }
  float2 res[16];
  fft512<true, false>(FFT_ARGS, sInRe[wv], sInIm[wv], sMidRe[wv], sMidIm[wv], res, lane);
  float2* dst = tmpb + (size_t)c * NXD;  // tmpb[ky][x]
#pragma unroll
  for (int t = 0; t < 2; ++t)
#pragma unroll
    for (int r = 0; r < 8; ++r) dst[xidx(r, t, lane)] = res[t * 8 + r];
}

// --- Inverse FFT over ky, real part, scale by zeta*dV/N^2 -> out[x][y].
__global__ __launch_bounds__(TPB) void k_ifft_cols(const float2* __restrict__ tmpb,
                                                   const float* __restrict__ h,
                                                   const float* __restrict__ tabs,
                                                   float* __restrict__ out) {
  FFT_SHARED
  load_tables(tabs, sTab);
  __syncthreads();
  const int lane = threadIdx.x & 31, wv = threadIdx.x >> 5;
  const int x = blockIdx.x * WPB + wv;
#pragma unroll
  for (int i = 0; i < 16; ++i) {
    int ky = lane + 32 * i;
    float2 v = tmpb[(size_t)ky * NXD + x];
    sInRe[wv][ky] = v.x;
    sInIm[wv][ky] = v.y;
  }
  float2 res[16];
  fft512<true, false>(FFT_ARGS, sInRe[wv], sInIm[wv], sMidRe[wv], sMidIm[wv], res, lane);
  const float zeta = 1.25663706212e-06f / (4.0f * 3.14159265358979f);
  const float scale = zeta * h[0] * h[1] * h[2] * (1.0f / (512.0f * 512.0f));
  float* dst = out + (size_t)x * NYD;
#pragma unroll
  for (int t = 0; t < 2; ++t)
#pragma unroll
    for (int r = 0; r < 8; ++r) dst[xidx(r, t, lane)] = res[t * 8 + r].x * scale;
}

__global__ void k_zero(float4* __restrict__ p, int n) {
  int i = blockIdx.x * blockDim.x + threadIdx.x;
  if (i < n) p[i] = make_float4(0.f, 0.f, 0.f, 0.f);
}

extern "C" void kernel_launch(void* const* d_in, const int* in_sizes, int n_in,
                              void* d_out, int out_size, void* d_ws, size_t ws_size,
                              hipStream_t stream) {
  (void)in_sizes; (void)n_in; (void)out_size; (void)ws_size;
  const float* M    = (const float*)d_in[0];
  const float* h    = (const float*)d_in[1];
  const float* dirs = (const float*)d_in[2];
  float* out = (float*)d_out;

  // Workspace layout (~50 MB total):
  //   tabs [3584] float            (16 KB slot, keeps rowM 16B-aligned)
  //   rowM [BZ][512][512] float2   (16 MB)
  //   rowP [BZ][512][512] float2   (16 MB)
  //   acc  [BZ][512][512] float2   (16 MB)  -- per-slot accumulators, no atomics
  //   tmpb [512][512]     float2   ( 2 MB)
  float*  tabs = (float*)d_ws;
  float2* rowM = (float2*)((char*)d_ws + 16384);
  float2* rowP = rowM + (size_t)BZ * NXD * NYD;
  float2* acc  = rowP + (size_t)BZ * NXD * NYD;
  float2* tmpb = acc  + (size_t)BZ * NXD * NYD;

  k_build_tables<<<dim3(1), TPB, 0, stream>>>(tabs);
  {
    int n = BZ * NXD * NYD / 2;  // float4 count for acc
    k_zero<<<dim3(n / 256), dim3(256), 0, stream>>>((float4*)acc, n);
  }

  for (int zb = 0; zb < NZD; zb += BZ) {
    dim3 g(NXD / WPB, BZ);
    k_fftrow_M<<<g, TPB, 0, stream>>>(M, tabs, rowM, zb);
    k_fftrow_P<<<g, TPB, 0, stream>>>(h, dirs, tabs, rowP, zb);
    k_fftcol_mac<<<g, TPB, 0, stream>>>(rowM, rowP, tabs, acc);
  }

  k_ifft_rows<<<dim3(NXD / WPB), TPB, 0, stream>>>(acc, tabs, tmpb);
  k_ifft_cols<<<dim3(NXD / WPB), TPB, 0, stream>>>(tmpb, h, tabs, out);
}